// mindssc_loss_46308337385674
// MI455X (gfx1250) — compile-verified
//
#include <hip/hip_runtime.h>
#include <cmath>

// ---------------------------------------------------------------------------
// MIND-SSC descriptor loss for MI455X (gfx1250, wave32).
// Gaussian smoothing is done on the matrix pipe: each wave computes a
// 16-line x 16-output tile as Out(16x16) = G(16x20) * X(20x16), split into
// five V_WMMA_F32_16X16X4_F32 accumulations (full f32 precision).
// ---------------------------------------------------------------------------

typedef __attribute__((ext_vector_type(2))) float v2f;
typedef __attribute__((ext_vector_type(8))) float v8f;

#define DD 160
#define HH 192
#define WW 160
#define DHW (DD * HH * WW)      // 4,915,200 = 19200 * 256
#define NB 1920                 // reduction blocks

__device__ __forceinline__ int iclamp(int v, int lo, int hi) {
    return v < lo ? lo : (v > hi ? hi : v);
}

// Gaussian tap by window distance d = (input window index) - (output index),
// valid taps at d in [0,4]; w0=w4, w1=w3 (symmetric), w2 = center.
__device__ __forceinline__ float wtap(int d, float w0, float w1, float w2) {
    float r = 0.f;
    r = (d == 0 || d == 4) ? w0 : r;
    r = (d == 1 || d == 3) ? w1 : r;
    r = (d == 2) ? w2 : r;
    return r;
}

// ---------------------------------------------------------------------------
// Pass 1: 12 squared-difference channels (face-neighbor pairs, edge clamp).
// ---------------------------------------------------------------------------
__global__ __launch_bounds__(256) void diff12_kernel(
    const float* __restrict__ img, float* __restrict__ ssd)
{
    int p = blockIdx.x * 256 + threadIdx.x;          // exact: DHW = 19200*256
    int w = p % WW;
    int t = p / WW;
    int h = t % HH;
    int d = t / HH;

    // 6 face-neighbor values (replicate padding == index clamp)
    float v0 = img[((iclamp(d - 1, 0, DD - 1)) * HH + h) * WW + w];   // (-1, 0, 0)
    float v1 = img[(d * HH + h) * WW + iclamp(w - 1, 0, WW - 1)];     // ( 0, 0,-1)
    float v2 = img[(d * HH + iclamp(h - 1, 0, HH - 1)) * WW + w];     // ( 0,-1, 0)
    float v3 = img[(d * HH + h) * WW + iclamp(w + 1, 0, WW - 1)];     // ( 0, 0,+1)
    float v4 = img[((iclamp(d + 1, 0, DD - 1)) * HH + h) * WW + w];   // (+1, 0, 0)
    float v5 = img[(d * HH + iclamp(h + 1, 0, HH - 1)) * WW + w];     // ( 0,+1, 0)

    float nv[6] = {v0, v1, v2, v3, v4, v5};
    const int pi[12] = {1, 2, 2, 3, 3, 4, 4, 4, 5, 5, 5, 5};
    const int pj[12] = {0, 0, 1, 0, 2, 1, 2, 3, 0, 1, 3, 4};
#pragma unroll
    for (int c = 0; c < 12; ++c) {
        float dv = nv[pi[c]] - nv[pj[c]];
        ssd[c * DHW + p] = dv * dv;
    }
}

// ---------------------------------------------------------------------------
// WMMA Gaussian smoothing along one axis.
//   S  : element stride along the smoothed axis
//   L  : length of the smoothed axis (multiple of 16)
//   NA : L/16 tiles along the axis
//   line -> memory base: base = (line / LD) * PS + (line % LD) * LS
// Each wave: 16 lines x 16 outputs, acc = sum_{c=0..4} A_c(16x4) * B_c(4x16).
// Lane layout (ISA 7.12.2, 32-bit): row/col = lane&15, K = v + 2*(lane>>4).
// ---------------------------------------------------------------------------
__global__ __launch_bounds__(256) void smooth_wmma_kernel(
    const float* __restrict__ in, float* __restrict__ out,
    int S, int L, int NA, int LD, int PS, int LS, int nwaves,
    float w0, float w1, float w2)
{
    int wid = blockIdx.x * 8 + (threadIdx.x >> 5);
    if (wid >= nwaves) return;                 // wave-uniform; EXEC stays full
    int lane = threadIdx.x & 31;
    int n = lane & 15;                          // B column / D column (line)
    int g = lane >> 4;                          // lane group

    int a_tile = wid % NA;
    int lb     = wid / NA;
    int x0     = a_tile * 16;                   // first output along axis
    int line   = lb * 16 + n;
    int base   = (line / LD) * PS + (line % LD) * LS;

    // Constant banded Gaussian weights: A[m][kg] = w[kg - m], kg in [0,20)
    int m = n;                                  // A row index = lane&15
    v2f A[5];
#pragma unroll
    for (int c = 0; c < 5; ++c) {
        int kg = 4 * c + 2 * g;                 // K = v + 2g, v is vector elem
        A[c].x = wtap(kg - m, w0, w1, w2);
        A[c].y = wtap(kg + 1 - m, w0, w1, w2);
    }

    v8f acc = {0.f, 0.f, 0.f, 0.f, 0.f, 0.f, 0.f, 0.f};
#pragma unroll
    for (int c = 0; c < 5; ++c) {
        int q  = x0 - 2 + 4 * c + 2 * g;        // input window position
        int i0 = iclamp(q, 0, L - 1);           // replicate padding
        int i1 = iclamp(q + 1, 0, L - 1);
        v2f B;
        B.x = in[base + i0 * S];
        B.y = in[base + i1 * S];
        // D = A x B + C   (v_wmma_f32_16x16x4_f32)
        acc = __builtin_amdgcn_wmma_f32_16x16x4_f32(
            false, A[c], false, B, (short)0, acc, false, false);
    }

    // D layout: VGPR r holds (m = r + 8g, col n) -> 8 consecutive outputs/lane
    if (S == 1) {
        float4* o = (float4*)(out + base + x0 + 8 * g);
        o[0] = make_float4(acc[0], acc[1], acc[2], acc[3]);
        o[1] = make_float4(acc[4], acc[5], acc[6], acc[7]);
    } else {
#pragma unroll
        for (int r = 0; r < 8; ++r)
            out[base + (x0 + 8 * g + r) * S] = acc[r];
    }
}

// ---------------------------------------------------------------------------
// mind_var partial sums: per voxel mean_c(ssd) - min_c(ssd), fixed-order tree.
// ---------------------------------------------------------------------------
__global__ __launch_bounds__(256) void reduce_mv_kernel(
    const float* __restrict__ ssd, float* __restrict__ partials)
{
    __shared__ float sm[256];
    int tid = threadIdx.x;
    float local = 0.f;
    for (int p = blockIdx.x * 256 + tid; p < DHW; p += NB * 256) {
        float mn = 3.4e38f, s = 0.f;
#pragma unroll
        for (int c = 0; c < 12; ++c) {
            float v = ssd[c * DHW + p];
            s += v;
            mn = fminf(mn, v);
        }
        local += s * (1.f / 12.f) - mn;
    }
    sm[tid] = local;
    __syncthreads();
    for (int o = 128; o > 0; o >>= 1) {
        if (tid < o) sm[tid] += sm[tid + o];
        __syncthreads();
    }
    if (tid == 0) partials[blockIdx.x] = sm[0];
}

__global__ __launch_bounds__(256) void finalize_sum_kernel(
    const float* __restrict__ partials, int n, float* __restrict__ dst, float scale)
{
    __shared__ float sm[256];
    int tid = threadIdx.x;
    float local = 0.f;
    for (int i = tid; i < n; i += 256) local += partials[i];
    sm[tid] = local;
    __syncthreads();
    for (int o = 128; o > 0; o >>= 1) {
        if (tid < o) sm[tid] += sm[tid + o];
        __syncthreads();
    }
    if (tid == 0) dst[0] = sm[0] * scale;
}

// ---------------------------------------------------------------------------
// Final loss: build both descriptors on the fly, accumulate (mx-my)^2.
// scal[0] = m_x, scal[1] = m_y (global means of mind_var).
// ---------------------------------------------------------------------------
__global__ __launch_bounds__(256) void loss_kernel(
    const float* __restrict__ sx, const float* __restrict__ sy,
    const float* __restrict__ scal, float* __restrict__ partials)
{
    __shared__ float sm[256];
    int tid = threadIdx.x;
    float m_x = scal[0], m_y = scal[1];
    float lox = 0.001f * m_x, hix = 1000.f * m_x;
    float loy = 0.001f * m_y, hiy = 1000.f * m_y;

    float local = 0.f;
    for (int p = blockIdx.x * 256 + tid; p < DHW; p += NB * 256) {
        float xv[12], yv[12];
        float mnx = 3.4e38f, sxx = 0.f, mny = 3.4e38f, syy = 0.f;
#pragma unroll
        for (int c = 0; c < 12; ++c) {
            xv[c] = sx[c * DHW + p];
            sxx += xv[c];
            mnx = fminf(mnx, xv[c]);
            yv[c] = sy[c * DHW + p];
            syy += yv[c];
            mny = fminf(mny, yv[c]);
        }
        float vx = sxx * (1.f / 12.f) - mnx;
        float vy = syy * (1.f / 12.f) - mny;
        vx = fminf(fmaxf(vx, lox), hix);
        vy = fminf(fmaxf(vy, loy), hiy);
        float rvx = 1.f / vx, rvy = 1.f / vy;
#pragma unroll
        for (int c = 0; c < 12; ++c) {
            float ex = expf(-(xv[c] - mnx) * rvx);
            float ey = expf(-(yv[c] - mny) * rvy);
            float dd = ex - ey;
            local += dd * dd;
        }
    }
    sm[tid] = local;
    __syncthreads();
    for (int o = 128; o > 0; o >>= 1) {
        if (tid < o) sm[tid] += sm[tid + o];
        __syncthreads();
    }
    if (tid == 0) partials[blockIdx.x] = sm[0];
}

// ---------------------------------------------------------------------------
// Host launch. Workspace layout (floats):
//   ssd_x[12*DHW] | ssd_y[12*DHW] | tmpA[DHW] | tmpB[DHW] | partials[NB] | scal[2]
//   total = 26*DHW + NB + 2  (~512 MB)
// ---------------------------------------------------------------------------
extern "C" void kernel_launch(void* const* d_in, const int* in_sizes, int n_in,
                              void* d_out, int out_size, void* d_ws, size_t ws_size,
                              hipStream_t stream) {
    (void)in_sizes; (void)n_in; (void)out_size; (void)ws_size;
    const float* imgs[2] = {(const float*)d_in[0], (const float*)d_in[1]};
    float* out = (float*)d_out;
    float* ws  = (float*)d_ws;

    float* ssd[2]   = {ws, ws + (size_t)12 * DHW};
    float* tmpA     = ws + (size_t)24 * DHW;
    float* tmpB     = ws + (size_t)25 * DHW;
    float* partials = ws + (size_t)26 * DHW;
    float* scal     = partials + NB;

    // Gaussian taps (sigma=0.8, N=5), computed in double like the reference.
    const double s2 = 2.0 * 0.8 * 0.8;
    const double e0 = std::exp(-4.0 / s2);
    const double e1 = std::exp(-1.0 / s2);
    const double sm = 1.0 + 2.0 * (e0 + e1);
    const float w0 = (float)(e0 / sm);
    const float w1 = (float)(e1 / sm);
    const float w2 = (float)(1.0 / sm);

    const int HW = HH * WW;
    const int NWAVES = 19200;   // per smoothing launch (exact for all 3 axes)

    for (int im = 0; im < 2; ++im) {
        diff12_kernel<<<DHW / 256, 256, 0, stream>>>(imgs[im], ssd[im]);
        for (int c = 0; c < 12; ++c) {
            float* plane = ssd[im] + (size_t)c * DHW;
            // axis D: S=HW, L=160, lines=(h,w): base = line
            smooth_wmma_kernel<<<NWAVES / 8, 256, 0, stream>>>(
                plane, tmpA, HW, DD, DD / 16, HW, 0, 1, NWAVES, w0, w1, w2);
            // axis H: S=W, L=192, lines=(d,w): base = (line/W)*HW + line%W
            smooth_wmma_kernel<<<NWAVES / 8, 256, 0, stream>>>(
                tmpA, tmpB, WW, HH, HH / 16, WW, HW, 1, NWAVES, w0, w1, w2);
            // axis W: S=1, L=160, lines=(d,h): base = line*W
            smooth_wmma_kernel<<<NWAVES / 8, 256, 0, stream>>>(
                tmpB, plane, 1, WW, WW / 16, DD * HH, 0, WW, NWAVES, w0, w1, w2);
        }
        reduce_mv_kernel<<<NB, 256, 0, stream>>>(ssd[im], partials);
        finalize_sum_kernel<<<1, 256, 0, stream>>>(partials, NB, scal + im,
                                                   1.0f / (float)DHW);
    }

    loss_kernel<<<NB, 256, 0, stream>>>(ssd[0], ssd[1], scal, partials);
    finalize_sum_kernel<<<1, 256, 0, stream>>>(partials, NB, out,
                                               1.0f / (12.0f * (float)DHW));
}